// HNGCL_83416854823259
// MI455X (gfx1250) — compile-verified
//
#include <hip/hip_runtime.h>
#include <hip/hip_bf16.h>
#include <math.h>

// HAN-style hetero graph conv + semantic attention for MI455X (gfx1250).
//
// Pipeline (all on `stream`):
//  1. zero workspace (degrees, agg, logits, sums)
//  2. degree counts per metapath (hw f32 atomics, L2-resident)
//  3. rsqrt(clamp(deg,1)) in place
//  4. edge scatter: agg[dst,m,:] += h[src,:]*rsqrt(out_deg)   (float4 + global_atomic_add_f32)
//  5. scale agg by rsqrt(in_deg) -> emb  [N,3,64]
//  6. WMMA attention: w = tanh(emb@W1+b1)@W2  via V_WMMA_F32_16X16X4_F32
//  7. per-metapath mean of w (LDS + global atomics)
//  8. softmax over 3 metapaths -> beta
//  9. out[n,:] = sum_m beta[m] * emb[n,m,:]

typedef __attribute__((ext_vector_type(2))) float v2f;
typedef __attribute__((ext_vector_type(8))) float v8f;

#define D_FEAT 64
#define H_HID  128
#define NMETA  3

// ---------------- 1) zero workspace ----------------
__global__ void __launch_bounds__(256) k_zero(float* __restrict__ p, long long n) {
  long long i = (long long)blockIdx.x * blockDim.x + threadIdx.x;
  long long stride = (long long)gridDim.x * blockDim.x;
  for (; i < n; i += stride) p[i] = 0.0f;
}

// ---------------- 2) degree counting ----------------
__global__ void __launch_bounds__(256) k_degrees(
    const long long* __restrict__ src, const long long* __restrict__ dst,
    float* __restrict__ deg_out, float* __restrict__ deg_in,
    long long E, long long Nn) {
  long long t = (long long)blockIdx.x * blockDim.x + threadIdx.x;
  if (t >= NMETA * E) return;
  long long m = t / E;
  long long s = src[t];
  long long d = dst[t];
  unsafeAtomicAdd(&deg_out[m * Nn + s], 1.0f);
  unsafeAtomicAdd(&deg_in [m * Nn + d], 1.0f);
}

// ---------------- 3) rsqrt(clamp(deg,1)) in place ----------------
__global__ void __launch_bounds__(256) k_rsqrt(float* __restrict__ deg, long long n) {
  long long t = (long long)blockIdx.x * blockDim.x + threadIdx.x;
  if (t >= n) return;
  float d = deg[t];
  deg[t] = rsqrtf(d < 1.0f ? 1.0f : d);
}

// ---------------- 4) edge scatter-add ----------------
// 16 threads per (metapath, edge); each thread handles a float4 chunk of the
// 64-float feature row. h/agg are L2-resident -> atomics resolve on-chip.
__global__ void __launch_bounds__(256) k_scatter(
    const float4* __restrict__ h4, const long long* __restrict__ src,
    const long long* __restrict__ dst, const float* __restrict__ rs_out,
    float* __restrict__ agg, long long E, long long Nn) {
  long long t = (long long)blockIdx.x * blockDim.x + threadIdx.x;
  long long total = NMETA * E * 16;
  if (t >= total) return;
  int lane16 = (int)(t & 15);
  long long eg = t >> 4;               // (m, e) flat index
  long long m = eg / E;
  long long s = src[eg];
  long long d = dst[eg];
  float rs = rs_out[m * Nn + s];
  float4 v = h4[s * (D_FEAT / 4) + lane16];
  float* dp = agg + (d * NMETA + m) * D_FEAT + lane16 * 4;
  unsafeAtomicAdd(dp + 0, v.x * rs);
  unsafeAtomicAdd(dp + 1, v.y * rs);
  unsafeAtomicAdd(dp + 2, v.z * rs);
  unsafeAtomicAdd(dp + 3, v.w * rs);
}

// ---------------- 5) scale by rsqrt(in_deg): agg -> emb ----------------
__global__ void __launch_bounds__(256) k_scale(
    float* __restrict__ agg, const float* __restrict__ rs_in,
    long long total, long long Nn) {
  long long t = (long long)blockIdx.x * blockDim.x + threadIdx.x;
  if (t >= total) return;
  long long n = t / (NMETA * D_FEAT);
  long long m = (t / D_FEAT) % NMETA;
  agg[t] *= rs_in[m * Nn + n];
}

// ---------------- 6) semantic attention GEMM via f32 WMMA ----------------
// Rows of emb ([N*3, 64]) x W1 ([64,128]); one wave owns a 16-row tile.
// 32-bit A 16x4 fragment: lane m = lane&15; VGPR0 K = 2*(lane>>4), VGPR1 K+1.
// B 4x16 fragment mirrors it with n = lane&15. C 16x16 f32: 8 VGPRs,
// M = v + 8*(lane>>4), N = lane&15.
__global__ void __launch_bounds__(256) k_attn_wmma(
    const float* __restrict__ emb, const float* __restrict__ W1,
    const float* __restrict__ b1, const float* __restrict__ W2,
    float* __restrict__ wl, int Mrows) {
  const int lane = threadIdx.x & 31;
  const int wave = threadIdx.x >> 5;
  const int tile = blockIdx.x * 8 + wave;   // 16-row tile
  const int row0 = tile * 16;
  if (row0 >= Mrows) return;

  const int n16  = lane & 15;
  const int half = lane >> 4;

  // Preload all 16 A fragments for this tile (K = 64 -> 16 steps of 4).
  v2f afrag[16];
  {
    int r = row0 + n16;
    if (r >= Mrows) r = Mrows - 1;        // clamp (tail tile only)
    const float* arow = emb + (long long)r * D_FEAT;
    #pragma unroll
    for (int ks = 0; ks < 16; ++ks) {
      int k = ks * 4 + 2 * half;
      v2f a; a.x = arow[k]; a.y = arow[k + 1];
      afrag[ks] = a;
    }
  }

  float p[8];
  #pragma unroll
  for (int v = 0; v < 8; ++v) p[v] = 0.0f;

  #pragma unroll
  for (int nt = 0; nt < H_HID / 16; ++nt) {        // 8 column tiles
    const int ncol = nt * 16 + n16;
    const float bias = b1[ncol];
    v8f acc;
    #pragma unroll
    for (int v = 0; v < 8; ++v) acc[v] = bias;     // C init = bias (same n per lane)
    #pragma unroll
    for (int ks = 0; ks < 16; ++ks) {
      v2f b;
      int kb = ks * 4 + 2 * half;
      b.x = W1[(long long)kb * H_HID + ncol];
      b.y = W1[(long long)(kb + 1) * H_HID + ncol];
      acc = __builtin_amdgcn_wmma_f32_16x16x4_f32(
          /*neg_a=*/false, afrag[ks], /*neg_b=*/false, b,
          /*c_mod=*/(short)0, acc, /*reuse_a=*/false, /*reuse_b=*/false);
    }
    // Fused epilogue: tanh then contract with W2 column (per-lane n fixed).
    const float w2 = W2[ncol];
    #pragma unroll
    for (int v = 0; v < 8; ++v) p[v] += tanhf(acc[v]) * w2;
  }

  // Reduce over the 16 lanes of each half (xor of bits 0..3 stays in-half).
  #pragma unroll
  for (int off = 8; off >= 1; off >>= 1) {
    #pragma unroll
    for (int v = 0; v < 8; ++v) p[v] += __shfl_xor(p[v], off, 32);
  }
  if (n16 == 0) {
    #pragma unroll
    for (int v = 0; v < 8; ++v) {
      int r = row0 + half * 8 + v;
      if (r < Mrows) wl[r] = p[v];
    }
  }
}

// ---------------- 7) per-metapath sum of logits ----------------
__global__ void __launch_bounds__(256) k_reduce(
    const float* __restrict__ wl, float* __restrict__ sums, long long Nn) {
  __shared__ float sm[NMETA];
  if (threadIdx.x < NMETA) sm[threadIdx.x] = 0.0f;
  __syncthreads();
  float l0 = 0.f, l1 = 0.f, l2 = 0.f;
  long long i = (long long)blockIdx.x * blockDim.x + threadIdx.x;
  long long stride = (long long)gridDim.x * blockDim.x;
  for (; i < Nn; i += stride) {
    const float* r = wl + i * NMETA;
    l0 += r[0]; l1 += r[1]; l2 += r[2];
  }
  atomicAdd(&sm[0], l0);
  atomicAdd(&sm[1], l1);
  atomicAdd(&sm[2], l2);
  __syncthreads();
  if (threadIdx.x < NMETA) unsafeAtomicAdd(&sums[threadIdx.x], sm[threadIdx.x]);
}

// ---------------- 8) softmax over metapaths ----------------
__global__ void k_softmax(const float* __restrict__ sums, float* __restrict__ beta,
                          float invN) {
  if (threadIdx.x == 0 && blockIdx.x == 0) {
    float a0 = sums[0] * invN, a1 = sums[1] * invN, a2 = sums[2] * invN;
    float mx = fmaxf(a0, fmaxf(a1, a2));
    float e0 = expf(a0 - mx), e1 = expf(a1 - mx), e2 = expf(a2 - mx);
    float inv = 1.0f / (e0 + e1 + e2);
    beta[0] = e0 * inv; beta[1] = e1 * inv; beta[2] = e2 * inv;
  }
}

// ---------------- 9) weighted sum over metapaths ----------------
__global__ void __launch_bounds__(256) k_final(
    const float* __restrict__ emb, const float* __restrict__ beta,
    float* __restrict__ out, long long n64) {
  long long t = (long long)blockIdx.x * blockDim.x + threadIdx.x;
  if (t >= n64) return;
  long long n = t >> 6;
  int d = (int)(t & 63);
  float b0 = beta[0], b1v = beta[1], b2 = beta[2];
  const float* e = emb + n * (NMETA * D_FEAT) + d;
  out[t] = b0 * e[0] + b1v * e[D_FEAT] + b2 * e[2 * D_FEAT];
}

extern "C" void kernel_launch(void* const* d_in, const int* in_sizes, int n_in,
                              void* d_out, int out_size, void* d_ws, size_t ws_size,
                              hipStream_t stream) {
  const float*     h   = (const float*)d_in[0];
  const long long* src = (const long long*)d_in[1];  // int64 [3,E]
  const long long* dst = (const long long*)d_in[2];  // int64 [3,E]
  const float*     W1  = (const float*)d_in[3];      // [64,128]
  const float*     b1  = (const float*)d_in[4];      // [128]
  const float*     W2  = (const float*)d_in[5];      // [128,1]
  float* out = (float*)d_out;

  const long long Nn = in_sizes[0] / D_FEAT;
  const long long E  = in_sizes[1] / NMETA;

  // Workspace layout (floats):
  float* ws      = (float*)d_ws;
  float* deg_out = ws;                          // 3*N
  float* deg_in  = deg_out + NMETA * Nn;        // 3*N
  float* agg     = deg_in  + NMETA * Nn;        // N*3*64 (becomes emb)
  float* wl      = agg + NMETA * Nn * D_FEAT;   // N*3 attention logits
  float* sums    = wl + NMETA * Nn;             // 3
  float* beta    = sums + NMETA;                // 3
  const long long zcount = (long long)(6 + NMETA * D_FEAT + NMETA) * Nn + NMETA;

  // 1) zero degrees + agg + wl + sums
  k_zero<<<16384, 256, 0, stream>>>(ws, zcount);

  // 2) degrees
  {
    long long tot = NMETA * E;
    k_degrees<<<(unsigned)((tot + 255) / 256), 256, 0, stream>>>(src, dst, deg_out, deg_in, E, Nn);
  }

  // 3) rsqrt over both degree arrays (contiguous 6*N)
  {
    long long tot = 2LL * NMETA * Nn;
    k_rsqrt<<<(unsigned)((tot + 255) / 256), 256, 0, stream>>>(deg_out, tot);
  }

  // 4) scatter
  {
    long long tot = NMETA * E * 16;
    k_scatter<<<(unsigned)((tot + 255) / 256), 256, 0, stream>>>(
        (const float4*)h, src, dst, deg_out, agg, E, Nn);
  }

  // 5) scale -> emb
  {
    long long tot = NMETA * Nn * D_FEAT;
    k_scale<<<(unsigned)((tot + 255) / 256), 256, 0, stream>>>(agg, deg_in, tot, Nn);
  }

  // 6) WMMA attention logits
  {
    int Mrows = (int)(NMETA * Nn);
    int tiles = (Mrows + 15) / 16;
    int blocks = (tiles + 7) / 8;               // 8 waves (tiles) per 256-thread block
    k_attn_wmma<<<blocks, 256, 0, stream>>>(agg, W1, b1, W2, wl, Mrows);
  }

  // 7) per-metapath sums
  k_reduce<<<1024, 256, 0, stream>>>(wl, sums, Nn);

  // 8) softmax -> beta
  k_softmax<<<1, 32, 0, stream>>>(sums, beta, 1.0f / (float)Nn);

  // 9) weighted sum -> out
  {
    long long tot = Nn * D_FEAT;
    k_final<<<(unsigned)((tot + 255) / 256), 256, 0, stream>>>(agg, beta, out, tot);
  }
}